// gConv2d_18751827215006
// MI455X (gfx1250) — compile-verified
//
#include <hip/hip_runtime.h>

typedef __attribute__((ext_vector_type(2))) float v2f;
typedef __attribute__((ext_vector_type(8))) float v8f;

#define O_CH 16
#define CIN  16
#define NG   12
#define MTOT 192   // O_CH*NG
#define CTOT 192   // CIN*NG
#define HOUT 64
#define WOUT 320
#define HP   66
#define WP   322
#define NB   8

// _HEX_POS order: (1,1),(0,0),(0,1),(1,2),(2,2),(2,1),(1,0); tap 7 = dummy (weight 0)
// dy = {1,0,0,1,2,2,1,1}, dx = {1,0,1,2,2,1,0,1} packed 2 bits per tap.
__device__ __forceinline__ int tap_dy(int t) {
  constexpr unsigned P = (1u<<0)|(0u<<2)|(0u<<4)|(1u<<6)|(2u<<8)|(2u<<10)|(1u<<12)|(1u<<14);
  return (int)((P >> (2*t)) & 3u);
}
__device__ __forceinline__ int tap_dx(int t) {
  constexpr unsigned P = (1u<<0)|(0u<<2)|(1u<<4)|(2u<<6)|(2u<<8)|(1u<<10)|(0u<<12)|(1u<<14);
  return (int)((P >> (2*t)) & 3u);
}

// LDS byte-offset of a pointer into a __shared__ object (for async-to-LDS ops).
// Cast goes pointer -> u64 -> u32 so the host pass (64-bit pointers) also
// type-checks; on device the addrspace(3) pointer is the 32-bit LDS offset.
__device__ __forceinline__ unsigned lds_off(const void* p) {
  typedef __attribute__((address_space(3))) const void* lds_ptr_t;
  return (unsigned)(unsigned long long)(lds_ptr_t)p;
}

// ---------------------------------------------------------------------------
// Phase 1: expand weight (O,Cin,12,7) into the WMMA A-operand layout
//   KwT[ci=192][tpair=4][m=192][2]   (tap 7 zero-padded)
// so that a wave's A fetch (lanes = consecutive m) is fully coalesced.
//   value = weight[o, c, perm[g,h], tap_src[g,t]],  m=o*12+g, ci=c*12+h,
//   t = tpair*2 + j.
// ---------------------------------------------------------------------------
__global__ void build_kw(const float* __restrict__ w, float* __restrict__ KwT) {
  int idx = blockIdx.x * blockDim.x + threadIdx.x;
  if (idx >= CTOT * 4 * MTOT * 2) return;
  int j  = idx & 1;
  int q  = idx >> 1;
  int m  = q % MTOT;
  int q2 = q / MTOT;
  int tp = q2 & 3;
  int ci = q2 >> 2;
  int t  = tp * 2 + j;
  float val = 0.0f;
  if (t < 7) {
    int o = m / NG, g = m % NG;
    int c = ci / NG, h = ci % NG;
    int f = g / 6, r = g % 6;
    int fh = h / 6, rh = h % 6;
    // gi = inv(g): f==0 -> (0,(6-r)%6), f==1 -> (1,r)
    int fgi = f;
    int rgi = f ? r : (6 - r) % 6;
    // perm = compose(gi, h)
    int pf = fgi ^ fh;
    int pr = (((rgi + (fgi == 0 ? rh : -rh)) % 6) + 6) % 6;
    int p  = pf * 6 + pr;
    // tap_src
    int src;
    if (t == 0) src = 0;
    else {
      int i = t - 1;
      int s = (f == 0) ? (((i - r) % 6 + 6) % 6) : (((r - i) % 6 + 6) % 6);
      src = 1 + s;
    }
    val = w[((o * CIN + c) * NG + p) * 7 + src];
  }
  KwT[idx] = val;
}

// ---------------------------------------------------------------------------
// Phase 2: implicit-GEMM hex conv with V_WMMA_F32_16X16X4_F32 (exact fp32).
// grid (5, 64, 8); block = 384 threads = 12 waves; wave w owns M-rows [16w,16w+16).
// Input is staged into a double-buffered LDS tile with GLOBAL_LOAD_ASYNC_TO_LDS
// (exactly 2 async instructions per wave per channel block -> precise
// s_wait_asynccnt double buffering; compute overlaps the next block's DMA).
// ---------------------------------------------------------------------------
__global__ __launch_bounds__(384)
void hexconv_wmma(const float* __restrict__ x, const float* __restrict__ KwT,
                  const float* __restrict__ bias, float* __restrict__ out) {
  // [buf][cc][dy][68]: 66 valid cols + pad; 68*4=272B and 204*4=816B are 16B-aligned.
  __shared__ __align__(16) float smem[2][8 * 204];

  const int b   = blockIdx.z;
  const int y   = blockIdx.y;
  const int x0  = blockIdx.x * 64;
  const int tid = threadIdx.x;
  const int wave = tid >> 5;
  const int lane = tid & 31;
  const int ln16 = lane & 15;
  const int hi   = lane >> 4;          // 0: lanes 0-15, 1: lanes 16-31
  const int mRow = wave * 16 + ln16;   // A-matrix row for this lane

  v8f acc0 = {}, acc1 = {}, acc2 = {}, acc3 = {};

  // Issue this thread's 2 async-to-LDS ops for channel block cb into buffer buf.
  auto stage = [&](int cb, int buf) {
    // b128: one 16B chunk per thread; 8ch * 3rows * 16chunks = 384 = blockDim.
    const int cc = tid / 48;
    const int rr = tid - cc * 48;
    const int dy = rr >> 4;
    const int q  = rr & 15;
    const float* g0 = &x[((b * CTOT + cb * 8 + cc) * HP + (y + dy)) * WP + x0 + 4 * q];
    unsigned l0 = lds_off(&smem[buf][cc * 204 + dy * 68 + 4 * q]);
    asm volatile("global_load_async_to_lds_b128 %0, %1, off"
                 :: "v"(l0), "v"(g0) : "memory");
    // b64 remainder (cols 64,65 of each of the 24 rows), replicated mod 24 so
    // every wave issues exactly the same number of async ops (same data -> benign).
    const int r2 = tid % 24;
    const int c2 = r2 / 3;
    const int d2 = r2 - c2 * 3;
    const float* g1 = &x[((b * CTOT + cb * 8 + c2) * HP + (y + d2)) * WP + x0 + 64];
    unsigned l1 = lds_off(&smem[buf][c2 * 204 + d2 * 68 + 64]);
    asm volatile("global_load_async_to_lds_b64 %0, %1, off"
                 :: "v"(l1), "v"(g1) : "memory");
  };

  stage(0, 0);

  for (int cb = 0; cb < 24; ++cb) {
    const int cur = cb & 1;
    if (cb + 1 < 24) {
      stage(cb + 1, cur ^ 1);                      // overwrites buffer consumed at cb-1
      asm volatile("s_wait_asynccnt 2" ::: "memory");  // my 2 loads for `cur` done
    } else {
      asm volatile("s_wait_asynccnt 0" ::: "memory");
    }
    __syncthreads();                                // all waves' `cur` data in LDS

    const float* sb = &smem[cur][0];
#pragma unroll
    for (int cc = 0; cc < 8; ++cc) {
      const int ci = cb * 8 + cc;
#pragma unroll
      for (int th = 0; th < 2; ++th) {
        // WMMA f32 16x16x4 operand layout:
        //  A: lane L -> M=L%16; VGPR0 = K {0|2}, VGPR1 = K {1|3} (by L/16)
        //  B: lane L -> N=L%16; VGPR0 = K {0|2}, VGPR1 = K {1|3}
        const int k0  = th * 4 + 2 * hi;       // tap feeding VGPR0 of this lane
        const int tp  = th * 2 + hi;           // tap-pair index into KwT
        const int dy0 = tap_dy(k0),     dx0 = tap_dx(k0);
        const int dy1 = tap_dy(k0 + 1), dx1 = tap_dx(k0 + 1);

        v2f a = *(const v2f*)(&KwT[(((ci * 4) + tp) * MTOT + mRow) * 2]);
        const float* s0 = &sb[cc * 204 + dy0 * 68 + dx0 + ln16];
        const float* s1 = &sb[cc * 204 + dy1 * 68 + dx1 + ln16];

        v2f bv;
        bv.x = s0[0];  bv.y = s1[0];
        acc0 = __builtin_amdgcn_wmma_f32_16x16x4_f32(false, a, false, bv,
                                                     (short)0, acc0, false, false);
        bv.x = s0[16]; bv.y = s1[16];
        acc1 = __builtin_amdgcn_wmma_f32_16x16x4_f32(false, a, false, bv,
                                                     (short)0, acc1, false, false);
        bv.x = s0[32]; bv.y = s1[32];
        acc2 = __builtin_amdgcn_wmma_f32_16x16x4_f32(false, a, false, bv,
                                                     (short)0, acc2, false, false);
        bv.x = s0[48]; bv.y = s1[48];
        acc3 = __builtin_amdgcn_wmma_f32_16x16x4_f32(false, a, false, bv,
                                                     (short)0, acc3, false, false);
      }
    }
    __syncthreads();   // everyone done reading `cur` before it is re-staged
  }

  // Epilogue: C/D layout VGPR v -> M = v + 8*hi (within the wave's 16-row tile),
  // N = lane%16. Write interior of padded output (shift by +1,+1) with bias.
  const int nb    = x0 + ln16 + 1;
  const int obase = (b * MTOT) * HP * WP + (y + 1) * WP;
#pragma unroll
  for (int v = 0; v < 8; ++v) {
    int m = wave * 16 + v + 8 * hi;
    float bvv = bias[m / NG];
    float* op = out + obase + m * HP * WP + nb;
    op[0]  = acc0[v] + bvv;
    op[16] = acc1[v] + bvv;
    op[32] = acc2[v] + bvv;
    op[48] = acc3[v] + bvv;
  }
}

// ---------------------------------------------------------------------------
// Phase 3: fill the 1-pixel border ring from the interior with the group
// rotation remap. Border reads are all interior (sy in [1,64], sx in [1,320]),
// so read/write sets are disjoint.
// ---------------------------------------------------------------------------
__global__ void fill_border(float* __restrict__ out) {
  const int total = NB * MTOT * (2 * WP + 2 * HOUT);  // 772 per (b,m)
  int idx = blockIdx.x * blockDim.x + threadIdx.x;
  if (idx >= total) return;
  int e  = idx % 772;
  int bm = idx / 772;
  int m  = bm % MTOT;
  int b  = bm / MTOT;

  int y, xx;
  if      (e < 322)  { y = 0;  xx = e; }
  else if (e < 644)  { y = 65; xx = e - 322; }
  else if (e < 708)  { xx = 0;   y = 1 + (e - 644); }
  else               { xx = 321; y = 1 + (e - 708); }

  int sy = (y == 0) ? 64 : (y == 65) ? 1 : y;
  int sx, dt;
  if      (xx == 0)   { sx = 320; dt = 1;  }
  else if (xx == 321) { sx = 1;   dt = -1; }
  else                { sx = xx;  dt = 0;  }

  int o = m / NG, t = m % NG;
  int f = t / 6, r = t % 6;
  int tt = f * 6 + (((r + dt) % 6) + 6) % 6;

  float v = out[((b * MTOT + o * NG + tt) * HP + sy) * WP + sx];
  out[((b * MTOT + m) * HP + y) * WP + xx] = v;
}

// ---------------------------------------------------------------------------
extern "C" void kernel_launch(void* const* d_in, const int* in_sizes, int n_in,
                              void* d_out, int out_size, void* d_ws, size_t ws_size,
                              hipStream_t stream) {
  (void)in_sizes; (void)n_in; (void)out_size; (void)ws_size;
  const float* x    = (const float*)d_in[0];  // (8, 192, 66, 322)
  const float* w    = (const float*)d_in[1];  // (16, 16, 12, 7)
  const float* bias = (const float*)d_in[2];  // (16,)
  float* out = (float*)d_out;                 // (8, 192, 66, 322)
  float* KwT = (float*)d_ws;                  // 192*4*192*2 floats = 1.18 MB

  {
    int n = CTOT * 4 * MTOT * 2;
    build_kw<<<(n + 255) / 256, 256, 0, stream>>>(w, KwT);
  }
  {
    dim3 grid(5, 64, 8);
    hexconv_wmma<<<grid, 384, 0, stream>>>(x, KwT, bias, out);
  }
  {
    int n = NB * MTOT * (2 * WP + 2 * HOUT);
    fill_border<<<(n + 255) / 256, 256, 0, stream>>>(out);
  }
}